// Net_6296422055966
// MI455X (gfx1250) — compile-verified
//
#include <hip/hip_runtime.h>

typedef __attribute__((ext_vector_type(2))) float v2f;
typedef __attribute__((ext_vector_type(8))) float v8f;

#define F_IN  256
#define F_MID 128
#define F_OUT 8

// ---------- degree / norm ----------
__global__ void k_fill1(float* __restrict__ deg, int n) {
  int i = blockIdx.x * blockDim.x + threadIdx.x;
  if (i < n) deg[i] = 1.0f;  // self loop contributes 1
}

__global__ void k_count(const int* __restrict__ dst, float* __restrict__ deg, int E) {
  int e = blockIdx.x * blockDim.x + threadIdx.x;
  if (e < E) atomicAdd(&deg[dst[e]], 1.0f);
}

__global__ void k_rsqrt(float* __restrict__ deg, int n) {
  int i = blockIdx.x * blockDim.x + threadIdx.x;
  if (i < n) deg[i] = rsqrtf(deg[i]);  // deg >= 1 always
}

// ---------- GEMM1: h0[N,128] = x[N,256] @ W1[256,128] via V_WMMA_F32_16X16X4_F32 ----------
// One wave computes a 16-row x 128-col output strip (8 WMMA accumulator tiles).
__global__ __launch_bounds__(256) void k_gemm1(const float* __restrict__ x,
                                               const float* __restrict__ W1,
                                               float* __restrict__ h0, int nstrips) {
  int strip = blockIdx.x * 8 + (threadIdx.x >> 5);
  if (strip >= nstrips) return;
  int lane = threadIdx.x & 31;
  int m    = lane & 15;     // A-row (lanes 0-15 and 16-31 both map M=0..15)
  int kh   = lane >> 4;     // K half: lanes 16-31 carry K=2,3
  const float* xrow = x + (size_t)(strip * 16 + m) * F_IN + 2 * kh;
  v8f acc[8] = {};
  for (int k = 0; k < F_IN; k += 4) {
    v2f a = *(const v2f*)(xrow + k);                       // A[m][k+2kh], A[m][k+2kh+1]
    const float* wrow = W1 + (size_t)(k + 2 * kh) * F_MID + m;
#pragma unroll
    for (int t = 0; t < 8; ++t) {
      v2f b;
      b.x = wrow[t * 16];                                  // B[k+2kh][n]
      b.y = wrow[t * 16 + F_MID];                          // B[k+2kh+1][n]
      acc[t] = __builtin_amdgcn_wmma_f32_16x16x4_f32(
          false, a, false, b, (short)0, acc[t], false, false);
    }
  }
  // C/D layout: lane n = lane&15, row = v + 8*kh
  int row_hi = strip * 16 + 8 * kh;
#pragma unroll
  for (int t = 0; t < 8; ++t)
#pragma unroll
    for (int v = 0; v < 8; ++v)
      h0[(size_t)(row_hi + v) * F_MID + t * 16 + m] = acc[t][v];
}

// ---------- GEMM2: h1[N,8] = h[N,128] @ W2[128,8] (N padded 8->16 with zero cols) ----------
__global__ __launch_bounds__(256) void k_gemm2(const float* __restrict__ h,
                                               const float* __restrict__ W2,
                                               float* __restrict__ h1, int nstrips) {
  int strip = blockIdx.x * 8 + (threadIdx.x >> 5);
  if (strip >= nstrips) return;
  int lane = threadIdx.x & 31;
  int n    = lane & 15;
  int kh   = lane >> 4;
  const float* hrow = h + (size_t)(strip * 16 + n) * F_MID + 2 * kh;
  v8f acc = {};
  for (int k = 0; k < F_MID; k += 4) {
    v2f a = *(const v2f*)(hrow + k);
    v2f b;
    b.x = (n < F_OUT) ? W2[(k + 2 * kh) * F_OUT + n] : 0.0f;
    b.y = (n < F_OUT) ? W2[(k + 2 * kh + 1) * F_OUT + n] : 0.0f;
    acc = __builtin_amdgcn_wmma_f32_16x16x4_f32(
        false, a, false, b, (short)0, acc, false, false);
  }
  if (n < F_OUT) {
    int row_hi = strip * 16 + 8 * kh;
#pragma unroll
    for (int v = 0; v < 8; ++v)
      h1[(size_t)(row_hi + v) * F_OUT + n] = acc[v];
  }
}

// ---------- self-loop seed: agg[i,f] = hsrc[i,f] * dinv[i]^2 ----------
__global__ void k_selfinit(const float* __restrict__ hsrc, const float* __restrict__ dinv,
                           float* __restrict__ agg, int total, int shift) {
  int idx = blockIdx.x * blockDim.x + threadIdx.x;
  if (idx < total) {
    float dv = dinv[idx >> shift];
    agg[idx] = hsrc[idx] * dv * dv;
  }
}

// ---------- edge scatter, width 128: one wave per edge, 4 floats per lane ----------
__global__ __launch_bounds__(256) void k_scatter128(const int* __restrict__ src,
                                                    const int* __restrict__ dst,
                                                    const float* __restrict__ dinv,
                                                    const float* __restrict__ h0,
                                                    float* __restrict__ agg, int E) {
  int e = blockIdx.x * 8 + (threadIdx.x >> 5);
  if (e >= E) return;
  int s = src[e], d = dst[e];
  float norm = dinv[s] * dinv[d];
  int f = (threadIdx.x & 31) * 4;
  const float4 hv = *(const float4*)(h0 + (size_t)s * F_MID + f);
  float* o = agg + (size_t)d * F_MID + f;
  atomicAdd(o + 0, hv.x * norm);
  atomicAdd(o + 1, hv.y * norm);
  atomicAdd(o + 2, hv.z * norm);
  atomicAdd(o + 3, hv.w * norm);
}

// ---------- edge scatter, width 8 ----------
__global__ void k_scatter8(const int* __restrict__ src, const int* __restrict__ dst,
                           const float* __restrict__ dinv, const float* __restrict__ h1,
                           float* __restrict__ agg, int E) {
  int idx = blockIdx.x * blockDim.x + threadIdx.x;
  if (idx >= E * 8) return;
  int e = idx >> 3, f = idx & 7;
  int s = src[e], d = dst[e];
  float norm = dinv[s] * dinv[d];
  atomicAdd(&agg[(size_t)d * F_OUT + f], h1[(size_t)s * F_OUT + f] * norm);
}

// ---------- epilogues ----------
__global__ void k_relu_bias(float* __restrict__ agg, const float* __restrict__ b, int total) {
  int idx = blockIdx.x * blockDim.x + threadIdx.x;
  if (idx < total) agg[idx] = fmaxf(agg[idx] + b[idx & (F_MID - 1)], 0.0f);
}

__global__ void k_bias8(float* __restrict__ agg, const float* __restrict__ b, int total) {
  int idx = blockIdx.x * blockDim.x + threadIdx.x;
  if (idx < total) agg[idx] += b[idx & (F_OUT - 1)];
}

// ---------- decoder: out[k] = dot8(z[a], z[b]) over concat(pos, neg) ----------
__global__ void k_edge_dot(const int* __restrict__ pos, const int* __restrict__ neg,
                           const float* __restrict__ z, float* __restrict__ out,
                           int P, int total) {
  int k = blockIdx.x * blockDim.x + threadIdx.x;
  if (k >= total) return;
  int a, b;
  if (k < P) { a = pos[k]; b = pos[P + k]; }
  else       { int kk = k - P; a = neg[kk]; b = neg[P + kk]; }
  const float4* za = (const float4*)(z + (size_t)a * F_OUT);
  const float4* zb = (const float4*)(z + (size_t)b * F_OUT);
  float4 a0 = za[0], a1 = za[1], b0 = zb[0], b1 = zb[1];
  out[k] = a0.x * b0.x + a0.y * b0.y + a0.z * b0.z + a0.w * b0.w +
           a1.x * b1.x + a1.y * b1.y + a1.z * b1.z + a1.w * b1.w;
}

extern "C" void kernel_launch(void* const* d_in, const int* in_sizes, int n_in,
                              void* d_out, int out_size, void* d_ws, size_t ws_size,
                              hipStream_t stream) {
  const float* x  = (const float*)d_in[0];
  const int*   ei = (const int*)d_in[1];
  const int*   pe = (const int*)d_in[2];
  const int*   ne = (const int*)d_in[3];
  const float* W1 = (const float*)d_in[4];
  const float* b1 = (const float*)d_in[5];
  const float* W2 = (const float*)d_in[6];
  const float* b2 = (const float*)d_in[7];
  float* out = (float*)d_out;

  const int N = in_sizes[0] / F_IN;   // 100000
  const int E = in_sizes[1] / 2;      // 1600000
  const int P = in_sizes[2] / 2;      // 200000
  const int* src = ei;
  const int* dst = ei + E;

  // workspace layout (floats), ~110 MB total
  float* deg  = (float*)d_ws;                     // N (reused as dinv in place)
  float* h0   = deg  + (((size_t)N + 255) & ~(size_t)255);
  float* agg1 = h0   + (size_t)N * F_MID;         // becomes h after relu+bias
  float* h1   = agg1 + (size_t)N * F_MID;
  float* agg2 = h1   + (size_t)N * F_OUT;         // becomes z after bias

  const int nstrips = (N + 15) / 16;
  const int TPB = 256;

  // degrees -> dinv
  k_fill1<<<(N + TPB - 1) / TPB, TPB, 0, stream>>>(deg, N);
  k_count<<<(E + TPB - 1) / TPB, TPB, 0, stream>>>(dst, deg, E);
  k_rsqrt<<<(N + TPB - 1) / TPB, TPB, 0, stream>>>(deg, N);

  // layer 1
  k_gemm1<<<(nstrips + 7) / 8, TPB, 0, stream>>>(x, W1, h0, nstrips);
  k_selfinit<<<((N * F_MID) + TPB - 1) / TPB, TPB, 0, stream>>>(h0, deg, agg1, N * F_MID, 7);
  k_scatter128<<<(E + 7) / 8, TPB, 0, stream>>>(src, dst, deg, h0, agg1, E);
  k_relu_bias<<<((N * F_MID) + TPB - 1) / TPB, TPB, 0, stream>>>(agg1, b1, N * F_MID);

  // layer 2
  k_gemm2<<<(nstrips + 7) / 8, TPB, 0, stream>>>(agg1, W2, h1, nstrips);
  k_selfinit<<<((N * F_OUT) + TPB - 1) / TPB, TPB, 0, stream>>>(h1, deg, agg2, N * F_OUT, 3);
  k_scatter8<<<((E * 8) + TPB - 1) / TPB, TPB, 0, stream>>>(src, dst, deg, h1, agg2, E);
  k_bias8<<<((N * F_OUT) + TPB - 1) / TPB, TPB, 0, stream>>>(agg2, b2, N * F_OUT);

  // decoder
  k_edge_dot<<<(out_size + TPB - 1) / TPB, TPB, 0, stream>>>(pe, ne, agg2, out, P, out_size);
}